// ModelNew_70428873719964
// MI455X (gfx1250) — compile-verified
//
#include <hip/hip_runtime.h>

// out[i,j] = A[i] * B[i,j]  (diag(A) @ B), fp32, N=M=8192 in the reference.
//
// Pure HBM-bandwidth kernel: 512 MiB moved, ~23 us at 23.3 TB/s on MI455X.
// - float4 (128-bit) coalesced NT loads/stores (TH_LOAD_NT / TH_STORE_NT):
//   B (256 MiB) exceeds the 192 MB L2 and is touched once; out is write-once.
// - Exact-tiling kernel has NO per-iteration bounds checks: all 4
//   global_load_b128 are issued back-to-back (LOADcnt=4 in flight) before the
//   first s_wait_loadcnt, then 8x v_pk_mul_f32, then 4 NT stores.
// - blockIdx.y = row, so A[row] is wave-uniform -> one s_load_b32 per block.

typedef float v4f __attribute__((ext_vector_type(4)));

#define TPB 256
#define V4_PER_THREAD 4

// Fast path: m_v4 is an exact multiple of TPB*V4_PER_THREAD (true for M=8192).
__global__ __launch_bounds__(TPB) void diag_scale_v4_exact_kernel(
    const float* __restrict__ A, const float* __restrict__ B,
    float* __restrict__ out, int m_v4 /* = M/4 */)
{
    const int row = blockIdx.y;
    const float a = A[row];                          // uniform -> s_load_b32
    const size_t base = (size_t)row * (size_t)m_v4;  // offset in float4 units

    const v4f* __restrict__ Bv = (const v4f*)B;
    v4f* __restrict__ Ov = (v4f*)out;

    const size_t c0 = base + (size_t)(blockIdx.x * (TPB * V4_PER_THREAD) + threadIdx.x);

    v4f b[V4_PER_THREAD];
#pragma unroll
    for (int u = 0; u < V4_PER_THREAD; ++u)          // 4 loads in flight
        b[u] = __builtin_nontemporal_load(&Bv[c0 + (size_t)(u * TPB)]);

#pragma unroll
    for (int u = 0; u < V4_PER_THREAD; ++u)          // drain: pk_mul + NT store
        __builtin_nontemporal_store(a * b[u], &Ov[c0 + (size_t)(u * TPB)]);
}

// Guarded path for ragged M (not hit for the 8192x8192 reference).
__global__ __launch_bounds__(TPB) void diag_scale_v4_guard_kernel(
    const float* __restrict__ A, const float* __restrict__ B,
    float* __restrict__ out, int m_v4)
{
    const int row = blockIdx.y;
    const float a = A[row];
    const size_t base = (size_t)row * (size_t)m_v4;

    const v4f* __restrict__ Bv = (const v4f*)B;
    v4f* __restrict__ Ov = (v4f*)out;

    const int c0 = blockIdx.x * (TPB * V4_PER_THREAD) + threadIdx.x;

#pragma unroll
    for (int u = 0; u < V4_PER_THREAD; ++u) {
        const int col = c0 + u * TPB;
        if (col < m_v4) {
            v4f b = __builtin_nontemporal_load(&Bv[base + col]);
            __builtin_nontemporal_store(a * b, &Ov[base + col]);
        }
    }
}

// Scalar fallback if M is not a multiple of 4.
__global__ __launch_bounds__(TPB) void diag_scale_scalar_kernel(
    const float* __restrict__ A, const float* __restrict__ B,
    float* __restrict__ out, int M)
{
    const int row = blockIdx.y;
    const float a = A[row];
    const size_t base = (size_t)row * (size_t)M;
    const int col = blockIdx.x * TPB + threadIdx.x;
    if (col < M) {
        float b = __builtin_nontemporal_load(&B[base + col]);
        __builtin_nontemporal_store(a * b, &out[base + col]);
    }
}

extern "C" void kernel_launch(void* const* d_in, const int* in_sizes, int n_in,
                              void* d_out, int out_size, void* d_ws, size_t ws_size,
                              hipStream_t stream) {
    const float* A = (const float*)d_in[0];
    const float* B = (const float*)d_in[1];
    float* out = (float*)d_out;

    const int N = in_sizes[0];                       // rows (len of A)
    const long long total = (long long)in_sizes[1];  // N*M elements of B
    const int M = (int)(total / N);                  // cols

    dim3 block(TPB);
    if ((M & 3) == 0) {
        const int m_v4 = M >> 2;
        const int tile = TPB * V4_PER_THREAD;
        if (m_v4 % tile == 0) {
            dim3 grid(m_v4 / tile, N);
            diag_scale_v4_exact_kernel<<<grid, block, 0, stream>>>(A, B, out, m_v4);
        } else {
            dim3 grid((m_v4 + tile - 1) / tile, N);
            diag_scale_v4_guard_kernel<<<grid, block, 0, stream>>>(A, B, out, m_v4);
        }
    } else {
        dim3 grid((M + TPB - 1) / TPB, N);
        diag_scale_scalar_kernel<<<grid, block, 0, stream>>>(A, B, out, M);
    }
}